// ChebNet_21328807592610
// MI455X (gfx1250) — compile-verified
//
#include <hip/hip_runtime.h>

#define FEAT   128
#define KDIM   384        // K * FEAT = 3 * 128
#define K4SEG  32         // k4 groups (of 4) per 128-wide segment
#define NOUTC  40

typedef __attribute__((ext_vector_type(2))) float v2f;
typedef __attribute__((ext_vector_type(8))) float v8f;

// ---------------------------------------------------------------------------
// elementwise helpers
// ---------------------------------------------------------------------------
__global__ void zero_f32(float* __restrict__ p, long long n) {
    long long i = (long long)blockIdx.x * blockDim.x + threadIdx.x;
    if (i < n) p[i] = 0.0f;
}

__global__ void deg_scatter(const int* __restrict__ dst, float* __restrict__ deg, int E) {
    int e = blockIdx.x * blockDim.x + threadIdx.x;
    if (e < E) unsafeAtomicAdd(&deg[dst[e]], 1.0f);
}

__global__ void deg_to_dinv(float* __restrict__ dinv, int n) {
    int i = blockIdx.x * blockDim.x + threadIdx.x;
    if (i < n) {
        float d = dinv[i];
        d = d < 1.0f ? 1.0f : d;
        dinv[i] = 1.0f / sqrtf(d);
    }
}

// acc -> X1 = -dinv * acc ; reset acc for the next scatter pass
__global__ void combine_x1(float* __restrict__ acc, float* __restrict__ X1,
                           const float* __restrict__ dinv, long long n) {
    long long i = (long long)blockIdx.x * blockDim.x + threadIdx.x;
    if (i < n) {
        float a = acc[i];
        acc[i] = 0.0f;
        X1[i] = -dinv[i >> 7] * a;   // i>>7 == i / FEAT
    }
}

// acc -> X2 = -2*dinv*acc - X0 ; reset acc
__global__ void combine_x2(float* __restrict__ acc, const float* __restrict__ X0,
                           float* __restrict__ X2, const float* __restrict__ dinv,
                           long long n) {
    long long i = (long long)blockIdx.x * blockDim.x + threadIdx.x;
    if (i < n) {
        float a = acc[i];
        acc[i] = 0.0f;
        X2[i] = -2.0f * dinv[i >> 7] * a - X0[i];
    }
}

// ---------------------------------------------------------------------------
// SpMM scatter: one wave (32 lanes) per edge, each lane a float4 feature chunk.
// Row gather is a fully coalesced 512B read of an L2-resident matrix; scatter
// uses native non-returning global_atomic_add_f32.
// acc[dst] += Y[src] * dinv[src]
// ---------------------------------------------------------------------------
__global__ void spmm_scatter(const float* __restrict__ Y, const int* __restrict__ src,
                             const int* __restrict__ dst, const float* __restrict__ dinv,
                             float* __restrict__ acc, int E) {
    long long gid = (long long)blockIdx.x * blockDim.x + threadIdx.x;
    int e = (int)(gid >> 5);
    if (e >= E) return;
    int lane = (int)(gid & 31);
    int s = src[e];
    int d = dst[e];
    float sc = dinv[s];
    const float4 y = *((const float4*)(Y + (long long)s * FEAT) + lane);
    float* ap = acc + (long long)d * FEAT + lane * 4;
    unsafeAtomicAdd(ap + 0, y.x * sc);
    unsafeAtomicAdd(ap + 1, y.y * sc);
    unsafeAtomicAdd(ap + 2, y.z * sc);
    unsafeAtomicAdd(ap + 3, y.w * sc);
}

// ---------------------------------------------------------------------------
// Weight pre-pack for the WMMA B-fragment layout (fp32 4x16 B tile):
//   Wp[(k4*2 + half)*ldw + col] = { W[4*k4 + 2*half][col], W[4*k4 + 2*half + 1][col] }
// Padded to ldw columns (zeros), so the GEMM inner loop has no bounds checks
// and each half-wave's B fetch is one contiguous 128B global_load_b64 run.
// ---------------------------------------------------------------------------
__global__ void pack_w(const float* __restrict__ W, float2* __restrict__ Wp,
                       int ncols, int ldw) {
    int i = blockIdx.x * blockDim.x + threadIdx.x;
    int total = (KDIM / 4) * 2 * ldw;
    if (i >= total) return;
    int col  = i % ldw;
    int half = (i / ldw) & 1;
    int k4   = i / (2 * ldw);
    int krow = 4 * k4 + 2 * half;
    float x = (col < ncols) ? W[(long long)krow * ncols + col] : 0.0f;
    float y = (col < ncols) ? W[(long long)(krow + 1) * ncols + col] : 0.0f;
    Wp[i] = make_float2(x, y);
}

// ---------------------------------------------------------------------------
// Fused ChebConv GEMM: H = relu([X0 | X1 | X2] @ W + b), fp32 WMMA 16x16x4.
// One wave computes a 16(M) x 64(N) slab: 4 accumulator tiles share each A
// fragment. K loop is split into 3 segments with compile-time base pointers
// (no flat addressing), all loads are unconditional b64.
//
// fp32 16x4 A layout: lanes 0-15 -> K = +0/+1, lanes 16-31 -> K = +2/+3
// fp32 4x16 B layout: symmetric; C/D: VGPR r = rows r / r+8.
// ---------------------------------------------------------------------------
__global__ void cheb_gemm_wmma(const float* __restrict__ X0, const float* __restrict__ X1,
                               const float* __restrict__ X2, const float2* __restrict__ Wp,
                               const float* __restrict__ bias, float* __restrict__ H,
                               int ldw, int ncolsOut) {
    const int lane    = threadIdx.x & 31;
    const int wave    = threadIdx.x >> 5;
    const int rowBase = blockIdx.x * 16;
    const int m       = lane & 15;        // row (A) / col (B) within tile
    const int half    = lane >> 4;        // 0: K+0/+1, 1: K+2/+3
    const int koff    = half << 1;
    const int colBase = wave * 64;

    v8f c[4];
    #pragma unroll
    for (int t = 0; t < 4; ++t)
        c[t] = (v8f){0.f, 0.f, 0.f, 0.f, 0.f, 0.f, 0.f, 0.f};

    #pragma unroll
    for (int seg = 0; seg < 3; ++seg) {
        const float* Xs = (seg == 0) ? X0 : (seg == 1) ? X1 : X2;
        const float* arow = Xs + (long long)(rowBase + m) * FEAT + koff;
        const float2* wseg = Wp + (long long)seg * (K4SEG * 2) * ldw
                                + (long long)half * ldw + colBase + m;
        #pragma unroll 8
        for (int k4 = 0; k4 < K4SEG; ++k4) {
            v2f a = *(const v2f*)(arow + k4 * 4);
            const float2* wrow = wseg + (long long)(k4 * 2) * ldw;
            #pragma unroll
            for (int t = 0; t < 4; ++t) {
                float2 bf = wrow[t * 16];
                v2f b;
                b.x = bf.x;
                b.y = bf.y;
                c[t] = __builtin_amdgcn_wmma_f32_16x16x4_f32(
                    /*neg_a=*/false, a, /*neg_b=*/false, b,
                    /*c_mod=*/(short)0, c[t], /*reuse_a=*/false, /*reuse_b=*/false);
            }
        }
    }

    // epilogue: bias + ReLU, masked store handles ragged output width (40)
    const int rowHi = half << 3;          // +0 or +8
    #pragma unroll
    for (int t = 0; t < 4; ++t) {
        const int col = colBase + t * 16 + m;
        if (col >= ncolsOut) continue;
        const float bb = bias[col];
        #pragma unroll
        for (int r = 0; r < 8; ++r) {
            const int row = rowBase + r + rowHi;
            float v = c[t][r] + bb;
            H[(long long)row * ncolsOut + col] = v > 0.0f ? v : 0.0f;
        }
    }
}

// ---------------------------------------------------------------------------
extern "C" void kernel_launch(void* const* d_in, const int* in_sizes, int n_in,
                              void* d_out, int out_size, void* d_ws, size_t ws_size,
                              hipStream_t stream) {
    const float* features = (const float*)d_in[0];
    const int*   src      = (const int*)d_in[1];
    const int*   dst      = (const int*)d_in[2];
    const float* W0       = (const float*)d_in[3];
    const float* b0       = (const float*)d_in[4];
    const float* Wh       = (const float*)d_in[5];
    const float* bh       = (const float*)d_in[6];
    const float* Wl       = (const float*)d_in[7];
    const float* bl       = (const float*)d_in[8];
    float*       out      = (float*)d_out;

    const int N = in_sizes[0] / FEAT;           // 100000
    const int E = in_sizes[1];                  // 1600000
    const long long NF = (long long)N * FEAT;   // 12.8M floats

    float*  acc  = (float*)d_ws;       // scatter accumulator
    float*  X1   = acc + NF;
    float*  X2   = X1 + NF;
    float*  Ha   = X2 + NF;
    float*  Hb   = Ha + NF;
    float*  dinv = Hb + NF;            // N floats (deg, then dinv)
    float2* Wp   = (float2*)(dinv + N);// packed weights, <= 96*2*128 float2

    const int B = 256;
    const int zgridNF = (int)((NF + B - 1) / B);
    const int gridN   = (N + B - 1) / B;
    const int gridE   = (E + B - 1) / B;
    const int gridEF  = (int)(((long long)E * 32 + B - 1) / B);
    const int gridM   = N / 16;        // 6250 row tiles (exact)

    // degree -> dinv
    zero_f32<<<zgridNF, B, 0, stream>>>(acc, NF);
    zero_f32<<<gridN, B, 0, stream>>>(dinv, (long long)N);
    deg_scatter<<<gridE, B, 0, stream>>>(dst, dinv, E);
    deg_to_dinv<<<gridN, B, 0, stream>>>(dinv, N);

    auto cheb = [&](const float* Xin, const float* W, const float* bias,
                    float* Hout, int ncols) {
        const int ldw = (ncols + 63) & ~63;           // 128 or 64
        // pack weights into fragment order (also zero-pads ragged widths)
        const int packTotal = (KDIM / 4) * 2 * ldw;
        pack_w<<<(packTotal + B - 1) / B, B, 0, stream>>>(W, Wp, ncols, ldw);
        // X1 = -spmm(Xin)
        spmm_scatter<<<gridEF, B, 0, stream>>>(Xin, src, dst, dinv, acc, E);
        combine_x1<<<zgridNF, B, 0, stream>>>(acc, X1, dinv, NF);
        // X2 = -2*spmm(X1) - Xin
        spmm_scatter<<<gridEF, B, 0, stream>>>(X1, src, dst, dinv, acc, E);
        combine_x2<<<zgridNF, B, 0, stream>>>(acc, Xin, X2, dinv, NF);
        // H = relu([Xin|X1|X2] @ W + b)
        const int wavesPerBlock = ldw / 64;           // 2 (hidden) or 1 (out)
        cheb_gemm_wmma<<<gridM, 32 * wavesPerBlock, 0, stream>>>(
            Xin, X1, X2, Wp, bias, Hout, ldw, ncols);
    };

    // input layer
    cheb(features, W0, b0, Ha, FEAT);
    const float* cur = Ha;
    // 5 hidden layers (ping-pong Ha/Hb)
    for (int i = 0; i < 5; ++i) {
        float* o = (i & 1) ? Ha : Hb;
        cheb(cur, Wh + (long long)i * KDIM * FEAT, bh + (long long)i * FEAT, o, FEAT);
        cur = o;
    }
    // output layer -> d_out
    cheb(cur, Wl, bl, out, NOUTC);
}